// Reduce_13735305413315
// MI455X (gfx1250) — compile-verified
//
#include <hip/hip_runtime.h>
#include <hip/hip_bf16.h>
#include <math.h>

// ---------------------------------------------------------------------------
// CDNA5 (gfx1250) implementation of the hierarchical reduce-net.
// Matrix work (the 12 conv1d layers, ~130 GFLOP fp32) runs on
// V_WMMA_F32_16X16X4_F32 (wave32, 16x16 D tile, K-step 4), wave tile
// 32 out-channels x 64 positions so each B-fragment feeds two WMMAs.
// All activations ping-pong through workspace buffers that fit in the
// 192 MB L2; sequence-length masks are recomputed statelessly from the
// original N via N_k = ((N-1)>>k)+1, so the launch is graph-capture safe.
// ---------------------------------------------------------------------------

#define HIDDEN 256
#define KW 5
#define CPAD 2          // conv padding (K-1)/2
#define LEAK 0.1f
#define EPSN 1e-5f
#define BATCH 16

typedef float v2f __attribute__((ext_vector_type(2)));
typedef float v8f __attribute__((ext_vector_type(8)));

// ---------------------------------------------------------------------------
// Implicit-GEMM conv1d via WMMA f32 16x16x4.
//   out[b,o,t] = bias[o] + sum_{k,i} w[o,i,k] * xin[b,i,stride*t+k-CPAD]
// Channel i==256 is the implicit depth channel (value logd, handled as a
// VALU rank-1 epilogue); input masked to p < N_in = ceil(N0/2^nshift).
// Grid: x = t-tiles(64), y = o-tiles(32), z = batch.  One wave per block.
// A-frag (16x4 f32): lane L holds W[o0+(L&15)][kk + 2*(L>>4) + {0,1}]
// B-frag (4x16 f32): lane L holds X[kk + 2*(L>>4) + {0,1}][t0 + (L&15)]
// (per CDNA5 ISA VGPR layout tables).
// ---------------------------------------------------------------------------
__global__ __launch_bounds__(32) void conv_wmma_kernel(
    const float* __restrict__ x, const float* __restrict__ w,
    const float* __restrict__ bias, float* __restrict__ out,
    const int* __restrict__ N0, int Lin, int Lout, int CO,
    int stride, int nshift, float logd)
{
  const int tb = blockIdx.x << 6;        // 64 positions per wave
  const int o0 = blockIdx.y << 5;        // 32 out-channels per wave
  const int b  = blockIdx.z;

  const int lane = threadIdx.x;
  const int m    = lane & 15;
  const int half = lane >> 4;

  const float* xb = x + (size_t)b * HIDDEN * Lin;
  int Nin = ((N0[b] - 1) >> nshift) + 1;
  if (Nin > Lin) Nin = Lin;

  v8f acc[8] = {};                       // [g*4 + s]: g = o-subtile, s = t-subtile

  const int WR = (HIDDEN + 1) * KW;      // 1285 reduction elems per out-channel

  for (int k = 0; k < KW; ++k) {
    // per-subtile input position + mask (fixed for the whole channel loop)
    float msk[4]; int pp[4];
#pragma unroll
    for (int s = 0; s < 4; ++s) {
      int t = tb + (s << 4) + m;
      int p = t * stride + k - CPAD;
      bool v = (p >= 0) && (p < Lin) && (p < Nin);
      msk[s] = v ? 1.0f : 0.0f;
      pp[s]  = v ? p : 0;
    }

    const float* wr0 = w + (size_t)(o0 + m) * WR + (size_t)(half << 1) * KW + k;
    const float* wr1 = wr0 + (size_t)16 * WR;
    const float* xc  = xb + (size_t)(half << 1) * Lin;

    // keep the weight stream hot (global_prefetch_b8)
    __builtin_prefetch(wr0 + 64 * KW, 0, 1);
    __builtin_prefetch(wr1 + 64 * KW, 0, 1);

#pragma unroll 2
    for (int ib = 0; ib < HIDDEN; ib += 4) {
      v2f av0, av1;
      av0.x = wr0[0];  av0.y = wr0[KW];   // W[o0+m,  ib+2h .. +1, k]
      av1.x = wr1[0];  av1.y = wr1[KW];   // W[o0+16+m, ...]
      wr0 += 4 * KW;  wr1 += 4 * KW;
#pragma unroll
      for (int s = 0; s < 4; ++s) {
        v2f bv;
        bv.x = msk[s] * xc[pp[s]];
        bv.y = msk[s] * xc[(size_t)Lin + pp[s]];
        acc[s] = __builtin_amdgcn_wmma_f32_16x16x4_f32(
            false, av0, false, bv, (short)0, acc[s], false, false);
        acc[4 + s] = __builtin_amdgcn_wmma_f32_16x16x4_f32(
            false, av1, false, bv, (short)0, acc[4 + s], false, false);
      }
      xc += (size_t)4 * Lin;
    }

    // depth channel (input channel 256): rank-1 update in VALU.
    // C/D layout: lane L element j -> (o = obase + j + 8*(L>>4), t = tb+16s+(L&15))
#pragma unroll
    for (int g = 0; g < 2; ++g) {
      float wd[8];
#pragma unroll
      for (int j = 0; j < 8; ++j)
        wd[j] = w[(size_t)(o0 + (g << 4) + j + (half << 3)) * WR
                  + (size_t)HIDDEN * KW + k];
#pragma unroll
      for (int s = 0; s < 4; ++s) {
        float dv = msk[s] * logd;
#pragma unroll
        for (int j = 0; j < 8; ++j) acc[(g << 2) + s][j] += wd[j] * dv;
      }
    }
  }

  // epilogue: bias + store
  float* ob = out + (size_t)b * CO * Lout;
#pragma unroll
  for (int s = 0; s < 4; ++s) {
    int t = tb + (s << 4) + m;
    if (t < Lout) {
#pragma unroll
      for (int g = 0; g < 2; ++g) {
#pragma unroll
        for (int j = 0; j < 8; ++j) {
          int o = o0 + (g << 4) + j + (half << 3);
          ob[(size_t)o * Lout + t] = acc[(g << 2) + s][j] + bias[o];
        }
      }
    }
  }
}

// ---------------------------------------------------------------------------
// Per-batch masked layernorm statistics: mu and rsqrt(var+eps) over
// (256 channels) x (t < Nn).  One 256-thread block per batch.
// ---------------------------------------------------------------------------
__global__ __launch_bounds__(256) void stats_kernel(
    const float* __restrict__ src, size_t batchStride, int Lout,
    const int* __restrict__ N0, int nshift,
    float* __restrict__ mu, float* __restrict__ rs)
{
  int b = blockIdx.x, tid = threadIdx.x;
  int Nn = ((N0[b] - 1) >> nshift) + 1;
  if (Nn > Lout) Nn = Lout;
  const float* p = src + (size_t)b * batchStride;

  float s = 0.f, s2 = 0.f;
  for (int ch = 0; ch < HIDDEN; ++ch) {
    const float* row = p + (size_t)ch * Lout;
    for (int t = tid; t < Nn; t += 256) {
      float v = row[t];
      s += v; s2 += v * v;
    }
  }
  __shared__ float sh[256], sh2[256];
  sh[tid] = s; sh2[tid] = s2;
  __syncthreads();
  for (int off = 128; off > 0; off >>= 1) {
    if (tid < off) { sh[tid] += sh[tid + off]; sh2[tid] += sh2[tid + off]; }
    __syncthreads();
  }
  if (tid == 0) {
    float inv = 1.0f / ((float)HIDDEN * (float)Nn);
    float mm  = sh[0] * inv;
    float var = sh2[0] * inv - mm * mm;
    mu[b] = mm;
    rs[b] = rsqrtf(var + EPSN);
  }
}

// ---------------------------------------------------------------------------
// xr = mask(t<Nn) * leaky( exp(lw)*(r - mu)*rs + bb )   (conv2 input prep)
// Lout is a power of two: lgL = log2(Lout).
// ---------------------------------------------------------------------------
__global__ __launch_bounds__(256) void xform_act_kernel(
    const float* __restrict__ src, size_t sBatchStride, int chOff,
    float* __restrict__ dst, int lgL, const int* __restrict__ N0, int nshift,
    const float* __restrict__ mu, const float* __restrict__ rs,
    const float* __restrict__ lw, const float* __restrict__ bb, int total)
{
  int idx = blockIdx.x * 256 + threadIdx.x;
  if (idx >= total) return;
  const int Lout = 1 << lgL;
  int t  = idx & (Lout - 1);
  int r  = idx >> lgL;
  int ch = r & (HIDDEN - 1);
  int b  = r >> 8;
  int Nn = ((N0[b] - 1) >> nshift) + 1;
  float A  = expf(lw[0]) * rs[b];
  float Cc = bb[0] - A * mu[b];
  float v = A * src[(size_t)b * sBatchStride + ((size_t)(chOff + ch) << lgL) + t] + Cc;
  v = fmaxf(v, 0.f) + LEAK * fminf(v, 0.f);
  dst[idx] = (t < Nn) ? v : 0.f;
}

// ---------------------------------------------------------------------------
// hnew = leaky( l + exp(lw)*(r2 - mu)*rs + bb )
// ---------------------------------------------------------------------------
__global__ __launch_bounds__(256) void combine_kernel(
    const float* __restrict__ lsrc, size_t lBatchStride,
    const float* __restrict__ r2, float* __restrict__ hnew, int lgL,
    const float* __restrict__ mu, const float* __restrict__ rs,
    const float* __restrict__ lw, const float* __restrict__ bb, int total)
{
  int idx = blockIdx.x * 256 + threadIdx.x;
  if (idx >= total) return;
  const int Lout = 1 << lgL;
  int t  = idx & (Lout - 1);
  int r  = idx >> lgL;
  int ch = r & (HIDDEN - 1);
  int b  = r >> 8;
  float A  = expf(lw[0]) * rs[b];
  float Cc = bb[0] - A * mu[b];
  float v = lsrc[(size_t)b * lBatchStride + ((size_t)ch << lgL) + t] + A * r2[idx] + Cc;
  hnew[idx] = fmaxf(v, 0.f) + LEAK * fminf(v, 0.f);
}

// ---------------------------------------------------------------------------
// Capture h[:, :, 0] for batches whose N collapses to <=1 exactly at `depth`.
// Stateless test: reduced-at-d  <=>  (N-1)>>(2(d+1))==0  and  (N-1)>>(2d)!=0
// ---------------------------------------------------------------------------
__global__ __launch_bounds__(256) void capture_kernel(
    const float* __restrict__ h, int Lout, const int* __restrict__ N0,
    int depth, float* __restrict__ cap)
{
  int b = blockIdx.x, ch = threadIdx.x;
  int nm1 = N0[b] - 1;
  bool now = ((nm1 >> (2 * (depth + 1))) == 0) &&
             (depth == 0 || ((nm1 >> (2 * depth)) != 0));
  if (now) cap[b * HIDDEN + ch] = h[((size_t)b * HIDDEN + ch) * Lout];
}

__device__ __forceinline__ int finish_of(int N)
{
  int nm1 = N - 1;
#pragma unroll
  for (int d = 0; d < 6; ++d)
    if ((nm1 >> (2 * (d + 1))) == 0) return d;
  return 5;
}

// Stable argsort by (finish*16 + batch): all keys distinct -> rank = #smaller.
__global__ __launch_bounds__(256) void order_kernel(
    const float* __restrict__ cap, const int* __restrict__ N0,
    float* __restrict__ out)
{
  int ch = threadIdx.x;
  for (int b = 0; b < BATCH; ++b) {
    int key = finish_of(N0[b]) * BATCH + b;
    int rank = 0;
    for (int b2 = 0; b2 < BATCH; ++b2)
      rank += ((finish_of(N0[b2]) * BATCH + b2) < key);
    out[rank * HIDDEN + ch] = cap[b * HIDDEN + ch];
  }
}

// ---------------------------------------------------------------------------
extern "C" void kernel_launch(void* const* d_in, const int* in_sizes, int n_in,
                              void* d_out, int out_size, void* d_ws, size_t ws_size,
                              hipStream_t stream)
{
  (void)in_sizes; (void)n_in; (void)out_size; (void)ws_size;

  const float* h0   = (const float*)d_in[0];   // (16,256,4096)
  const int*   N0   = (const int*)  d_in[1];   // (16,)
  const float* c1w  = (const float*)d_in[2];   // (2,512,257,5)
  const float* c1b  = (const float*)d_in[3];   // (2,512)
  const float* c2w  = (const float*)d_in[4];   // (2,256,257,5)
  const float* c2b  = (const float*)d_in[5];   // (2,256)
  const float* n1lw = (const float*)d_in[6];
  const float* n1b  = (const float*)d_in[7];
  const float* n2lw = (const float*)d_in[8];
  const float* n2b  = (const float*)d_in[9];
  float* out = (float*)d_out;

  const size_t LMAX = 2048;
  float* ws = (float*)d_ws;
  size_t off = 0;
  float* hp0 = ws + off; off += (size_t)BATCH * HIDDEN * LMAX;     // 33.5 MB
  float* hp1 = ws + off; off += (size_t)BATCH * HIDDEN * LMAX;     // 33.5 MB
  float* lr  = ws + off; off += (size_t)BATCH * 2 * HIDDEN * LMAX; // 67 MB
  float* r2  = ws + off; off += (size_t)BATCH * HIDDEN * LMAX;     // 33.5 MB
  float* mu1 = ws + off; off += BATCH;
  float* rs1 = ws + off; off += BATCH;
  float* mu2 = ws + off; off += BATCH;
  float* rs2 = ws + off; off += BATCH;
  float* cap = ws + off; off += (size_t)BATCH * HIDDEN;

  const int Ld[13] = {4096, 2048, 1024, 512, 256, 128, 64, 32, 16, 8, 4, 2, 1};

  for (int j = 0; j < 12; ++j) {
    const int   layer = j & 1;
    const int   depth = j >> 1;
    const int   Lin   = Ld[j];
    const int   Lo    = Ld[j + 1];
    const int   lgL   = __builtin_ctz(Lo);       // Lout is a power of two
    const float logd  = log1pf((float)depth);

    const float* hin = (j == 0) ? h0 : ((j & 1) ? hp0 : hp1);
    float* hnew = (j & 1) ? hp1 : hp0;
    float* xr   = (j & 1) ? hp0 : hp1;   // dead input buffer, safe to reuse

    // conv1: 257 -> 512 channels, stride 2, mask with N_j
    {
      dim3 grid((Lo + 63) / 64, 512 / 32, BATCH);
      conv_wmma_kernel<<<grid, 32, 0, stream>>>(
          hin, c1w + (size_t)layer * 512 * 257 * KW, c1b + layer * 512, lr,
          N0, Lin, Lo, 512, /*stride=*/2, /*nshift=*/j, logd);
    }
    // layernorm-1 stats on r half (channels 256..511), mask with N_{j+1}
    stats_kernel<<<BATCH, 256, 0, stream>>>(
        lr + (size_t)HIDDEN * Lo, (size_t)2 * HIDDEN * Lo, Lo, N0, j + 1, mu1, rs1);
    // normalize + leaky + mask -> conv2 input
    {
      int total = BATCH * HIDDEN * Lo;
      xform_act_kernel<<<(total + 255) / 256, 256, 0, stream>>>(
          lr, (size_t)2 * HIDDEN * Lo, HIDDEN, xr, lgL, N0, j + 1,
          mu1, rs1, n1lw + layer, n1b + layer, total);
    }
    // conv2: 257 -> 256 channels, stride 1, mask with N_{j+1}
    {
      dim3 grid((Lo + 63) / 64, 256 / 32, BATCH);
      conv_wmma_kernel<<<grid, 32, 0, stream>>>(
          xr, c2w + (size_t)layer * 256 * 257 * KW, c2b + layer * 256, r2,
          N0, Lo, Lo, 256, /*stride=*/1, /*nshift=*/j + 1, logd);
    }
    // layernorm-2 stats
    stats_kernel<<<BATCH, 256, 0, stream>>>(
        r2, (size_t)HIDDEN * Lo, Lo, N0, j + 1, mu2, rs2);
    // residual combine + leaky
    {
      int total = BATCH * HIDDEN * Lo;
      combine_kernel<<<(total + 255) / 256, 256, 0, stream>>>(
          lr, (size_t)2 * HIDDEN * Lo, r2, hnew, lgL,
          mu2, rs2, n2lw + layer, n2b + layer, total);
    }
    // after each depth (second block), capture finished batches
    if (j & 1)
      capture_kernel<<<BATCH, 256, 0, stream>>>(hnew, Lo, N0, depth, cap);
  }

  order_kernel<<<1, 256, 0, stream>>>(cap, N0, out);
}